// TTTCausalConv1d_18451179503928
// MI455X (gfx1250) — compile-verified
//
#include <hip/hip_runtime.h>
#include <stdint.h>

// Problem constants (reference: B=4, S=4096, H=2048, K=4)
#define B_ 4
#define S_ 4096
#define H_ 2048
#define K_ 4
#define TOK 8                    // output tokens per block
#define HALO (K_ - 1)            // causal left halo
#define ROWS (TOK + HALO)        // 11 rows staged in LDS
#define NTHREADS 256
#define NWAVES (NTHREADS / 32)
#define EPS 1e-6f

typedef __attribute__((ext_vector_type(4))) float        v4f;
typedef __attribute__((ext_vector_type(4))) unsigned int u32x4;
typedef __attribute__((ext_vector_type(8))) int          i32x8;
typedef __attribute__((ext_vector_type(4))) int          i32x4;

#if defined(__has_builtin)
#if __has_builtin(__builtin_amdgcn_tensor_load_to_lds) && \
    __has_builtin(__builtin_amdgcn_s_wait_tensorcnt)
#define USE_TDM 1
#endif
#endif

__global__ __launch_bounds__(NTHREADS)
void fused_rmsnorm_dwconv1d(const float* __restrict__ x,   // [B,S,H]
                            const float* __restrict__ nw,  // [H] norm weight
                            const float* __restrict__ cw,  // [H,1,K] conv weight
                            const float* __restrict__ cb,  // [H] conv bias
                            float* __restrict__ y)         // [B,S,H]
{
    __shared__ float s_invrms[ROWS + 1];
    __shared__ float s_tile[ROWS * H_];          // 11 * 2048 * 4B = 88 KiB

    const int tid  = threadIdx.x;
    const int bIdx = (int)blockIdx.x / (S_ / TOK);
    const int ti   = (int)blockIdx.x % (S_ / TOK);
    const int s0   = ti * TOK;
    const int zrows = (s0 == 0) ? HALO : 0;      // rows of causal zero-pad
    const int nrows = ROWS - zrows;              // rows DMA'd from global

#ifdef USE_TDM
    // ---- Tensor Data Mover: one 2D tile load (nrows x 2048 fp32) ----
    if (tid < 32) {                              // wave 0 only (TDM ignores EXEC)
        const float* gsrc = x + ((size_t)bIdx * S_ + (size_t)(s0 - HALO + zrows)) * H_;
        // LDS generic-pointer low 32 bits == workgroup-relative LDS byte offset
        unsigned lds_off = (unsigned)(unsigned long long)(uintptr_t)&s_tile[zrows * H_];
        unsigned long long ga = (unsigned long long)(uintptr_t)gsrc;

        // D# group0: count=1 | lds_addr | global_addr[56:0] | type=2
        u32x4 g0 = { 1u,
                     lds_off,
                     (unsigned)(ga & 0xffffffffull),
                     (unsigned)((ga >> 32) & 0x1ffffffull) | (2u << 30) };
        // D# group1: data_size=4B; tensor_dim0=2048; tensor_dim1=nrows;
        //            tile_dim0=2048; tile_dim1=nrows; dim0_stride=2048
        i32x8 g1 = { (int)(2u << 16),                 // [17:16] data_size=2 (4B)
                     (int)((unsigned)H_ << 16),       // tensor_dim0 low16 @ [63:48]
                     (int)((unsigned)nrows << 16),    // tensor_dim1 low16 @ [95:80]
                     (int)((unsigned)H_ << 16),       // tile_dim0 @ [127:112]
                     (int)(unsigned)nrows,            // tile_dim1 @ [143:128]
                     (int)H_,                         // tensor_dim0_stride low32
                     0, 0 };
        i32x4 g2 = { 1, 0, 0, 0 };                    // tensor_dim2=1 (tile_dim2=0)
        i32x4 g3 = { 0, 0, 0, 0 };
#if __has_include(<hip/amd_detail/amd_gfx1250_TDM.h>)
#warning CDNA5_PROBE: TDM path, 6-arg builtin (therock headers present)
        i32x8 g4 = { 0, 0, 0, 0, 0, 0, 0, 0 };
        __builtin_amdgcn_tensor_load_to_lds(g0, g1, g2, g3, g4, 0);
#else
#warning CDNA5_PROBE: TDM path, 5-arg builtin (ROCm lane)
        __builtin_amdgcn_tensor_load_to_lds(g0, g1, g2, g3, 0);
#endif
    }
    // zero-fill causal pad rows (disjoint from TDM destination rows)
    for (int i = tid; i < zrows * H_; i += NTHREADS) s_tile[i] = 0.0f;
    if (tid < 32) __builtin_amdgcn_s_wait_tensorcnt(0);
#else
#warning CDNA5_PROBE: fallback path, no tensor_load_to_lds builtin
    // ---- Fallback: vectorized global -> LDS copy ----
    for (int i = tid; i < ROWS * (H_ / 4); i += NTHREADS) {
        int row = i >> 9;                 // H_/4 = 512 float4 per row
        int col = (i & 511) << 2;
        int gs  = s0 - HALO + row;
        v4f v = { 0.f, 0.f, 0.f, 0.f };
        if (gs >= 0)
            v = *(const v4f*)(x + ((size_t)bIdx * S_ + gs) * H_ + col);
        *(v4f*)&s_tile[row * H_ + col] = v;
    }
#endif
    __syncthreads();

    // ---- Per-row RMS: one wave32 per row, b128 LDS reads + shfl_xor tree ----
    const int lane = tid & 31;
    const int wv   = tid >> 5;
    for (int r = wv; r < ROWS; r += NWAVES) {
        const v4f* rp = (const v4f*)&s_tile[r * H_];
        float acc = 0.f;
#pragma unroll
        for (int j = 0; j < H_ / (32 * 4); ++j) {     // 16 x ds_load_b128
            v4f v = rp[lane + 32 * j];
            acc = fmaf(v.x, v.x, acc);
            acc = fmaf(v.y, v.y, acc);
            acc = fmaf(v.z, v.z, acc);
            acc = fmaf(v.w, v.w, acc);
        }
#pragma unroll
        for (int off = 16; off > 0; off >>= 1) acc += __shfl_xor(acc, off, 32);
        if (lane == 0) s_invrms[r] = rsqrtf(acc * (1.0f / H_) + EPS);
    }
    __syncthreads();

    // preload all row scale factors into registers (broadcast, conflict-free)
    float ir[ROWS];
#pragma unroll
    for (int r = 0; r < ROWS; ++r) ir[r] = s_invrms[r];

    // ---- Depthwise causal conv: 4 consecutive channels per lane (b128) ----
#pragma unroll
    for (int g = 0; g < H_ / (NTHREADS * 4); ++g) {   // 2 passes over H
        const int hb = g * (NTHREADS * 4) + tid * 4;
        // conv_weight [H,1,K]: 4 contiguous float4 loads, transpose to taps
        v4f wc0 = *(const v4f*)(cw + (size_t)(hb + 0) * K_);
        v4f wc1 = *(const v4f*)(cw + (size_t)(hb + 1) * K_);
        v4f wc2 = *(const v4f*)(cw + (size_t)(hb + 2) * K_);
        v4f wc3 = *(const v4f*)(cw + (size_t)(hb + 3) * K_);
        v4f wt0 = { wc0.x, wc1.x, wc2.x, wc3.x };
        v4f wt1 = { wc0.y, wc1.y, wc2.y, wc3.y };
        v4f wt2 = { wc0.z, wc1.z, wc2.z, wc3.z };
        v4f wt3 = { wc0.w, wc1.w, wc2.w, wc3.w };
        v4f bv  = *(const v4f*)(cb + hb);
        v4f nwv = *(const v4f*)(nw + hb);

        // rolling 4-row normalized window: 1 new LDS b128 load per token
        v4f x0 = (*(const v4f*)&s_tile[0 * H_ + hb]) * ir[0];
        v4f x1 = (*(const v4f*)&s_tile[1 * H_ + hb]) * ir[1];
        v4f x2 = (*(const v4f*)&s_tile[2 * H_ + hb]) * ir[2];
        float* yp = y + ((size_t)bIdx * S_ + s0) * H_ + hb;
#pragma unroll
        for (int t = 0; t < TOK; ++t) {
            v4f x3 = (*(const v4f*)&s_tile[(t + HALO) * H_ + hb]) * ir[t + HALO];
            v4f acc = wt0 * x0 + wt1 * x1 + wt2 * x2 + wt3 * x3;
            v4f out = bv + nwv * acc;
            // write-once 128 MiB stream: non-temporal b128 store keeps L2 for input
            __builtin_nontemporal_store(out, (v4f*)(yp + (size_t)t * H_));
            x0 = x1; x1 = x2; x2 = x3;
        }
    }
}

extern "C" void kernel_launch(void* const* d_in, const int* in_sizes, int n_in,
                              void* d_out, int out_size, void* d_ws, size_t ws_size,
                              hipStream_t stream) {
    (void)in_sizes; (void)n_in; (void)out_size; (void)d_ws; (void)ws_size;
    const float* x  = (const float*)d_in[0];   // hidden_states [B,S,H] fp32
    const float* nw = (const float*)d_in[1];   // norm_weight [H]
    const float* cw = (const float*)d_in[2];   // conv_weight [H,1,K]
    const float* cb = (const float*)d_in[3];   // conv_bias [H]
    float* y = (float*)d_out;

    dim3 grid(B_ * (S_ / TOK));                // 2048 blocks
    fused_rmsnorm_dwconv1d<<<grid, NTHREADS, 0, stream>>>(x, nw, cw, cb, y);
}